// UniGATConv_21131239096601
// MI455X (gfx1250) — compile-verified
//
#include <hip/hip_runtime.h>
#include <math.h>

#define C_IN   256
#define C_OUT  128
#define NEG_SLOPE 0.2f

typedef float v2f __attribute__((ext_vector_type(2)));
typedef float v8f __attribute__((ext_vector_type(8)));

// ---------------------------------------------------------------------------
// Kernel 1: Xw = X * W^T + b   (fp32 WMMA, 16x16x4)
// Block = 256 threads = 8 waves; wave w -> N-tile w (C_OUT = 8*16).
// blockIdx.x -> M-tile (N_V = 3125*16, exact).  EXEC all-ones throughout.
// ---------------------------------------------------------------------------
__global__ __launch_bounds__(256) void k_gemm_theta(
    const float* __restrict__ X, const float* __restrict__ W,
    const float* __restrict__ b, float* __restrict__ Xw) {
  const int wave = threadIdx.x >> 5;       // 0..7 -> n-tile
  const int lane = threadIdx.x & 31;
  const int half = lane >> 4;              // 0 or 1
  const int l    = lane & 15;
  const int m0   = blockIdx.x * 16;
  const int n0   = wave * 16;

  const float* __restrict__ arow = X + (size_t)(m0 + l) * C_IN + half * 2;
  const float* __restrict__ brow = W + (size_t)(n0 + l) * C_IN + half * 2;

  v8f c = {};
#pragma unroll 8
  for (int k = 0; k < C_IN; k += 4) {
    v2f a  = *(const v2f*)(arow + k);
    v2f bb = *(const v2f*)(brow + k);
    c = __builtin_amdgcn_wmma_f32_16x16x4_f32(
        /*neg_a=*/false, a, /*neg_b=*/false, bb,
        /*c_mod=*/(short)0, c, /*reuse_a=*/false, /*reuse_b=*/false);
  }

  const float bias = b[n0 + l];
#pragma unroll
  for (int r = 0; r < 8; ++r) {
    const int row = m0 + r + half * 8;     // C/D layout: VGPR r -> M=r / M=r+8
    Xw[(size_t)row * C_OUT + n0 + l] = c[r] + bias;
  }
}

// ---------------------------------------------------------------------------
// Kernel 2: per-segment incidence counts (int atomics only)
// ---------------------------------------------------------------------------
__global__ __launch_bounds__(256) void k_count(
    const int* __restrict__ pv, const int* __restrict__ pe,
    int* __restrict__ cnt_v, int* __restrict__ cnt_e, int npairs) {
  const int p = blockIdx.x * blockDim.x + threadIdx.x;
  if (p >= npairs) return;
  atomicAdd(cnt_v + pv[p], 1);
  atomicAdd(cnt_e + pe[p], 1);
}

// ---------------------------------------------------------------------------
// Kernel 3: single-block exclusive scan (chunk sums -> Hillis-Steele -> emit)
// ---------------------------------------------------------------------------
__global__ __launch_bounds__(1024) void k_scan_excl(
    const int* __restrict__ cnt, int* __restrict__ off, int n) {
  __shared__ int sums[1024];
  const int t = threadIdx.x;
  const int L = (n + 1023) / 1024;
  const int beg = t * L;
  const int end = min(beg + L, n);
  int s = 0;
  for (int i = beg; i < end; ++i) s += cnt[i];
  sums[t] = s;
  __syncthreads();
  for (int d = 1; d < 1024; d <<= 1) {
    int x = (t >= d) ? sums[t - d] : 0;
    __syncthreads();
    sums[t] += x;
    __syncthreads();
  }
  int run = sums[t] - s;                   // exclusive prefix of this chunk
  for (int i = beg; i < end; ++i) { off[i] = run; run += cnt[i]; }
  if (t == 1023) off[n] = sums[1023];
}

// ---------------------------------------------------------------------------
// Kernel 4: fill CSR adjacency (slot via int atomic cursor)
// csr_v[slot] = incident hyperedge; csr_e[slot] = incident vertex
// ---------------------------------------------------------------------------
__global__ __launch_bounds__(256) void k_fill(
    const int* __restrict__ pv, const int* __restrict__ pe,
    int* __restrict__ cur_v, int* __restrict__ cur_e,
    int* __restrict__ csr_v, int* __restrict__ csr_e, int npairs) {
  const int p = blockIdx.x * blockDim.x + threadIdx.x;
  if (p >= npairs) return;
  const int v = pv[p];
  const int e = pe[p];
  csr_v[atomicAdd(cur_v + v, 1)] = e;
  csr_e[atomicAdd(cur_e + e, 1)] = v;
}

// s(p) = clip(leakyrelu(alpha_e), 0.001, 5.0)  -- always in [0.001, 5]
__device__ __forceinline__ float pair_score(float a) {
  float s = (a >= 0.0f) ? a : NEG_SLOPE * a;
  return fminf(fmaxf(s, 0.001f), 5.0f);
}

// ---------------------------------------------------------------------------
// Kernel 5: edge gather.  One wave per hyperedge, 4 channels per lane.
// Y[e] = mean of member Xw rows;  alpha[e] = Y[e] . a_e  (wave32 xor-reduce)
// ---------------------------------------------------------------------------
__global__ __launch_bounds__(256) void k_edge_gather(
    const float* __restrict__ Xw, const int* __restrict__ off_e,
    const int* __restrict__ csr_e, const float* __restrict__ ae,
    float* __restrict__ Y, float* __restrict__ alpha, int ne) {
  const int e    = blockIdx.x * 8 + (threadIdx.x >> 5);
  const int lane = threadIdx.x & 31;
  if (e >= ne) return;
  const int beg = off_e[e];
  const int end = off_e[e + 1];
  float4 acc = make_float4(0.f, 0.f, 0.f, 0.f);
  for (int idx = beg; idx < end; ++idx) {
    const int v = csr_e[idx];
    if (idx + 1 < end)   // hide the index->row dependent-load latency
      __builtin_prefetch(Xw + (size_t)csr_e[idx + 1] * C_OUT + lane * 4, 0, 3);
    const float4 x = ((const float4*)(Xw + (size_t)v * C_OUT))[lane];
    acc.x += x.x; acc.y += x.y; acc.z += x.z; acc.w += x.w;
  }
  const float inv = 1.0f / fmaxf((float)(end - beg), 1.0f);
  float4 y = make_float4(acc.x * inv, acc.y * inv, acc.z * inv, acc.w * inv);
  ((float4*)(Y + (size_t)e * C_OUT))[lane] = y;
  const float4 a = ((const float4*)ae)[lane];
  float d = y.x * a.x + y.y * a.y + y.z * a.z + y.w * a.w;
#pragma unroll
  for (int o = 16; o > 0; o >>= 1) d += __shfl_xor(d, o, 32);
  if (lane == 0) alpha[e] = d;
}

// ---------------------------------------------------------------------------
// Kernel 6: vertex gather (fused segment-softmax + weighted sum + ELU).
// One wave per vertex.  Pass 1: m = max s.  Pass 2: num += w*Y[e], den += w.
// out[v] = ELU(num/den).  No atomics; each output row written exactly once.
// ---------------------------------------------------------------------------
__global__ __launch_bounds__(256) void k_vertex_gather(
    const float* __restrict__ Y, const float* __restrict__ alpha,
    const int* __restrict__ off_v, const int* __restrict__ csr_v,
    float* __restrict__ out, int nv) {
  const int v    = blockIdx.x * 8 + (threadIdx.x >> 5);
  const int lane = threadIdx.x & 31;
  if (v >= nv) return;
  const int beg = off_v[v];
  const int end = off_v[v + 1];

  float m = 0.0f;                          // scores are >= 0.001
  for (int idx = beg; idx < end; ++idx)
    m = fmaxf(m, pair_score(alpha[csr_v[idx]]));

  float4 num = make_float4(0.f, 0.f, 0.f, 0.f);
  float  den = 0.0f;
  for (int idx = beg; idx < end; ++idx) {
    const int e = csr_v[idx];
    if (idx + 1 < end)
      __builtin_prefetch(Y + (size_t)csr_v[idx + 1] * C_OUT + lane * 4, 0, 3);
    const float w = __expf(pair_score(alpha[e]) - m);
    den += w;
    const float4 y = ((const float4*)(Y + (size_t)e * C_OUT))[lane];
    num.x += w * y.x; num.y += w * y.y; num.z += w * y.z; num.w += w * y.w;
  }
  const float invden = (den > 0.0f) ? (1.0f / den) : 0.0f;
  float4 o = make_float4(num.x * invden, num.y * invden,
                         num.z * invden, num.w * invden);
  o.x = (o.x > 0.f) ? o.x : expm1f(o.x);
  o.y = (o.y > 0.f) ? o.y : expm1f(o.y);
  o.z = (o.z > 0.f) ? o.z : expm1f(o.z);
  o.w = (o.w > 0.f) ? o.w : expm1f(o.w);
  ((float4*)(out + (size_t)v * C_OUT))[lane] = o;   // empty segment -> 0 == ref
}

// ---------------------------------------------------------------------------
extern "C" void kernel_launch(void* const* d_in, const int* in_sizes, int n_in,
                              void* d_out, int out_size, void* d_ws, size_t ws_size,
                              hipStream_t stream) {
  const float* X   = (const float*)d_in[0];
  const float* W   = (const float*)d_in[1];
  const float* b   = (const float*)d_in[2];
  const float* a_e = (const float*)d_in[3];
  const int* pv    = (const int*)d_in[4];
  const int* pe    = (const int*)d_in[5];

  const int N_V     = in_sizes[0] / C_IN;     // 50000
  const int N_PAIRS = in_sizes[4];            // 800000
  const int N_E     = 10000;                  // fixed by the reference

  // ---- carve scratch from d_ws (256B-aligned regions) ----
  size_t off = 0;
  auto carve = [&](size_t bytes) {
    void* p = (char*)d_ws + off;
    off += (bytes + 255) & ~(size_t)255;
    return p;
  };
  float* Xw    = (float*)carve((size_t)N_V * C_OUT * sizeof(float));   // 25.6 MB
  float* Y     = (float*)carve((size_t)N_E * C_OUT * sizeof(float));   //  5.1 MB
  float* alpha = (float*)carve((size_t)N_E * sizeof(float));
  int* cnt_v   = (int*)carve((size_t)N_V * sizeof(int));
  int* cnt_e   = (int*)carve((size_t)N_E * sizeof(int));
  int* off_v   = (int*)carve((size_t)(N_V + 1) * sizeof(int));
  int* off_e   = (int*)carve((size_t)(N_E + 1) * sizeof(int));
  int* cur_v   = (int*)carve((size_t)N_V * sizeof(int));
  int* cur_e   = (int*)carve((size_t)N_E * sizeof(int));
  int* csr_v   = (int*)carve((size_t)N_PAIRS * sizeof(int));           //  3.2 MB
  int* csr_e   = (int*)carve((size_t)N_PAIRS * sizeof(int));           //  3.2 MB
  (void)ws_size;

  hipMemsetAsync(cnt_v, 0, (size_t)N_V * sizeof(int), stream);
  hipMemsetAsync(cnt_e, 0, (size_t)N_E * sizeof(int), stream);

  // 1) GEMM (WMMA): one wave per 16x16 tile; 8 waves/block = all 8 N-tiles.
  k_gemm_theta<<<N_V / 16, 256, 0, stream>>>(X, W, b, Xw);

  // 2) build CSR: count -> scan -> fill
  k_count<<<(N_PAIRS + 255) / 256, 256, 0, stream>>>(pv, pe, cnt_v, cnt_e, N_PAIRS);
  k_scan_excl<<<1, 1024, 0, stream>>>(cnt_v, off_v, N_V);
  k_scan_excl<<<1, 1024, 0, stream>>>(cnt_e, off_e, N_E);
  hipMemcpyAsync(cur_v, off_v, (size_t)N_V * sizeof(int),
                 hipMemcpyDeviceToDevice, stream);
  hipMemcpyAsync(cur_e, off_e, (size_t)N_E * sizeof(int),
                 hipMemcpyDeviceToDevice, stream);
  k_fill<<<(N_PAIRS + 255) / 256, 256, 0, stream>>>(
      pv, pe, cur_v, cur_e, csr_v, csr_e, N_PAIRS);

  // 3) v2e mean aggregation + attention logits (gather, no float atomics)
  k_edge_gather<<<(N_E + 7) / 8, 256, 0, stream>>>(
      Xw, off_e, csr_e, a_e, Y, alpha, N_E);

  // 4) fused segment-softmax + weighted e2v aggregation + ELU
  k_vertex_gather<<<(N_V + 7) / 8, 256, 0, stream>>>(
      Y, alpha, off_v, csr_v, (float*)d_out, N_V);
}